// Decoder_19224273616935
// MI455X (gfx1250) — compile-verified
//
#include <hip/hip_runtime.h>
#include <math.h>

// ---------------------------------------------------------------------------
// Problem constants
// ---------------------------------------------------------------------------
#define VOCAB      500000
#define EMBED_DIM  128
#define HIDDEN     128
#define STEPS      64
#define M_TILES    (VOCAB / 16)      // 31250 tiles of 16 rows
#define BSWZ_FLOATS 8192             // 32 chunks * 4 ntiles * 32 lanes * 2
#define GMAX_BYTE_OFF 32768          // after Bswz (32KB) in d_ws

typedef float v2f __attribute__((ext_vector_type(2)));
typedef float v8f __attribute__((ext_vector_type(8)));

__device__ __forceinline__ float sigmoidf_(float x) {
    return 1.0f / (1.0f + __expf(-x));
}

// ---------------------------------------------------------------------------
// Kernel 1: serial LSTM (single workgroup, 512 threads = one gate row each).
//   gates = A1 * x + A2 * h + bias2,  A1 = w_ih[:, :128], A2 = w_hh,
//   bias2 = w_ih[:, 128:] @ inp + b_ih + b_hh   (inp is loop-invariant!)
// Writes cs -> d_out[0..8191], c-vectors in WMMA-B swizzled layout -> bswz,
// and zero-inits the 64 global argmax slots.
// ---------------------------------------------------------------------------
__global__ __launch_bounds__(512) void lstm_kernel(
    const float* __restrict__ inp, const float* __restrict__ embed,
    const float* __restrict__ w_ih, const float* __restrict__ w_hh,
    const float* __restrict__ b_ih, const float* __restrict__ b_hh,
    float* __restrict__ cs_out, float* __restrict__ bswz,
    unsigned long long* __restrict__ gmax)
{
    __shared__ float xsh[128];   // current input x (starts as embed[<start>])
    __shared__ float hsh[128];
    __shared__ float csh[128];
    __shared__ float insh[128];
    __shared__ float act[512];

    const int g = threadIdx.x;

    if (g < 64)  gmax[g] = 0ULL;
    if (g < 128) {
        insh[g] = inp[g];
        xsh[g]  = embed[g];      // START_IDX == 0 -> row 0
        hsh[g]  = 0.0f;
        csh[g]  = 0.0f;
    }

    // w_hh row -> registers (128 VGPRs/thread)
    float4 a2v[32];
    const float4* whhp = (const float4*)(w_hh + (size_t)g * HIDDEN);
    #pragma unroll
    for (int j = 0; j < 32; ++j) a2v[j] = whhp[j];

    __syncthreads();

    // bias2 precompute (uses insh)
    const float4* a1p = (const float4*)(w_ih + (size_t)g * 256);
    float b2 = b_ih[g] + b_hh[g];
    {
        const float4* in4 = (const float4*)insh;
        #pragma unroll
        for (int j = 0; j < 32; ++j) {
            float4 w = a1p[32 + j];   // columns 128..255 act on inp
            float4 v = in4[j];
            b2 = fmaf(w.x, v.x, fmaf(w.y, v.y, fmaf(w.z, v.z, fmaf(w.w, v.w, b2))));
        }
    }

    const float4* xs4 = (const float4*)xsh;
    const float4* hs4 = (const float4*)hsh;

    for (int t = 0; t < STEPS; ++t) {
        __syncthreads();              // x/h/c updates from prev step visible
        float acc = b2;
        #pragma unroll
        for (int j = 0; j < 32; ++j) {
            float4 w = a1p[j];        // streamed from L2 (512KB resident)
            float4 v = xs4[j];
            acc = fmaf(w.x, v.x, fmaf(w.y, v.y, fmaf(w.z, v.z, fmaf(w.w, v.w, acc))));
        }
        #pragma unroll
        for (int j = 0; j < 32; ++j) {
            float4 w = a2v[j];        // register-resident w_hh row
            float4 v = hs4[j];
            acc = fmaf(w.x, v.x, fmaf(w.y, v.y, fmaf(w.z, v.z, fmaf(w.w, v.w, acc))));
        }
        act[g] = acc;
        __syncthreads();

        if (g < 128) {
            float iv = sigmoidf_(act[g]);
            float fv = sigmoidf_(act[128 + g]);
            float gv = tanhf(act[256 + g]);
            float ov = sigmoidf_(act[384 + g]);
            float cn = fmaf(fv, csh[g], iv * gv);
            float hn = ov * tanhf(cn);
            csh[g] = cn;
            hsh[g] = hn;
            xsh[g] = cn;              // next x is the cell state
            cs_out[t * HIDDEN + g] = cn;

            // WMMA B layout: VGPR r, lanes 0-15 -> K = 4c+r, lanes 16-31 -> K = 4c+r+2
            // element (t, k): chunk = k>>2, rm = k&3, hi = rm>>1, r = rm&1
            int chunk = g >> 2;
            int rm    = g & 3;
            int hi    = rm >> 1;
            int r     = rm & 1;
            int nt    = t >> 4;
            int lane  = (hi << 4) | (t & 15);
            bswz[(size_t)(((chunk * 4 + nt) * 32 + lane) * 2 + r)] = cn;
        }
    }
}

// ---------------------------------------------------------------------------
// Kernel 2: embed[500000x128] x C^T[128x64] via V_WMMA_F32_16X16X4_F32,
// fused 1/w_norm scaling + per-column argmax.
// One wave per 16-row M-tile (grid-stride). B (32KB) staged in LDS and
// re-read per chunk (memory clobber stops the compiler register-hoisting
// all 256 B values). A loads are batched per tile for 32-deep MLP.
// ---------------------------------------------------------------------------
__global__ __launch_bounds__(256) void gemm_argmax_kernel(
    const float* __restrict__ embed, const float* __restrict__ bswz,
    unsigned long long* __restrict__ gmax)
{
    __shared__ float Bls[BSWZ_FLOATS];            // 32 KB
    __shared__ unsigned long long amax[64];

    const int tid = threadIdx.x;

    // stage B into LDS (float4 coalesced)
    for (int i = tid; i < BSWZ_FLOATS / 4; i += 256)
        ((float4*)Bls)[i] = ((const float4*)bswz)[i];
    if (tid < 64) amax[tid] = 0ULL;
    __syncthreads();

    const int lane = tid & 31;
    const int wave = tid >> 5;
    const int lo   = lane & 15;     // row-in-tile (A) / column-in-tile (B)
    const int hi   = lane >> 4;     // K-half selector

    const int waveGlobal = blockIdx.x * 8 + wave;
    const int nWaves     = gridDim.x * 8;

    float best[4] = {-INFINITY, -INFINITY, -INFINITY, -INFINITY};
    int   bidx[4] = {0, 0, 0, 0};

    const v2f* Bv = (const v2f*)Bls;

    for (int tile = waveGlobal; tile < M_TILES; tile += nWaves) {
        // keep B in LDS (block register-hoisting of all 256 B values)
        asm volatile("" ::: "memory");

        const int m0 = tile << 4;
        const float* aptr = embed + (size_t)(m0 + lo) * EMBED_DIM + 2 * hi;

        // prefetch next tile's rows into cache while we compute
        if (tile + nWaves < M_TILES) {
            const float* pn = embed + (size_t)(m0 + nWaves * 16 + lo) * EMBED_DIM;
            __builtin_prefetch(pn, 0, 3);
            __builtin_prefetch(pn + 64, 0, 3);
        }

        // ---- batch ALL A loads for this tile: 32 outstanding b64 loads ----
        v2f areg[32];
        #pragma unroll
        for (int c = 0; c < 32; ++c)
            areg[c] = *(const v2f*)(aptr + 4 * c);   // A: K = 4c+2hi, 4c+2hi+1

        v8f acc[4] = {};
        float ss = 0.0f;

        #pragma unroll
        for (int c = 0; c < 32; ++c) {
            v2f a = areg[c];
            ss = fmaf(a.x, a.x, fmaf(a.y, a.y, ss));    // free row-norm partial
            const v2f* bp = Bv + (size_t)c * 128 + lane;
            #pragma unroll
            for (int nt = 0; nt < 4; ++nt)
                acc[nt] = __builtin_amdgcn_wmma_f32_16x16x4_f32(
                    false, a, false, bp[nt * 32], (short)0, acc[nt], false, false);
        }

        // full row sum-of-squares: combine the two K-halves (lane L <-> L+16)
        float tot = ss + __shfl_xor(ss, 16, 32);
        float inv = 1.0f / fmaxf(__fsqrt_rn(tot), 1e-8f);   // c_norm drops out of argmax

        // broadcast the 8 row scales this lane needs (rows r + 8*hi)
        float sc[8];
        #pragma unroll
        for (int r = 0; r < 8; ++r)
            sc[r] = __shfl(inv, r + hi * 8, 32);

        // D layout: VGPR r -> row (r + 8*hi), column = lo  => scale + argmax
        #pragma unroll
        for (int nt = 0; nt < 4; ++nt) {
            #pragma unroll
            for (int r = 0; r < 8; ++r) {
                float v = acc[nt][r] * sc[r];
                if (v > best[nt]) { best[nt] = v; bidx[nt] = m0 + r + hi * 8; }
            }
        }
    }

    // combine the two lane-halves of each column, then LDS u64 atomic-max
    #pragma unroll
    for (int nt = 0; nt < 4; ++nt) {
        float v = best[nt];
        int   m = bidx[nt];
        float ov = __shfl_xor(v, 16, 32);
        int   om = __shfl_xor(m, 16, 32);
        if (ov > v || (ov == v && om < m)) { v = ov; m = om; }
        if (hi == 0) {
            unsigned key = __float_as_uint(v);
            key = (key & 0x80000000u) ? ~key : (key | 0x80000000u);  // monotone map
            unsigned long long packed =
                ((unsigned long long)key << 32) | (unsigned)(~(unsigned)m); // ties -> lowest idx
            atomicMax(&amax[nt * 16 + lo], packed);
        }
    }
    __syncthreads();
    if (tid < 64) atomicMax(&gmax[tid], amax[tid]);   // one global u64 max per column/block
}

// ---------------------------------------------------------------------------
// Kernel 3: unpack winners -> d_out[8192 + t] as floats
// ---------------------------------------------------------------------------
__global__ void finalize_kernel(const unsigned long long* __restrict__ gmax,
                                float* __restrict__ out)
{
    int t = threadIdx.x;
    if (t < STEPS) {
        unsigned idx = ~(unsigned)(gmax[t] & 0xFFFFFFFFull);
        out[STEPS * HIDDEN + t] = (float)idx;
    }
}

// ---------------------------------------------------------------------------
extern "C" void kernel_launch(void* const* d_in, const int* in_sizes, int n_in,
                              void* d_out, int out_size, void* d_ws, size_t ws_size,
                              hipStream_t stream) {
    const float* inp   = (const float*)d_in[0];
    const float* embed = (const float*)d_in[1];
    const float* w_ih  = (const float*)d_in[2];
    const float* w_hh  = (const float*)d_in[3];
    const float* b_ih  = (const float*)d_in[4];
    const float* b_hh  = (const float*)d_in[5];
    float* out = (float*)d_out;

    float* bswz = (float*)d_ws;
    unsigned long long* gmax =
        (unsigned long long*)((char*)d_ws + GMAX_BYTE_OFF);

    // Phase 1: serial LSTM (also zero-inits gmax, writes cs + swizzled B)
    lstm_kernel<<<1, 512, 0, stream>>>(inp, embed, w_ih, w_hh, b_ih, b_hh,
                                       out, bswz, gmax);

    // Phase 2: one bandwidth-bound pass over embed (256 MB), f32 WMMA
    gemm_argmax_kernel<<<512, 256, 0, stream>>>(embed, bswz, gmax);

    // Phase 3: decode argmax winners
    finalize_kernel<<<1, 64, 0, stream>>>(gmax, out);
}